// WaveletTransform_30459908063858
// MI455X (gfx1250) — compile-verified
//
#include <hip/hip_runtime.h>
#include <hip/hip_bf16.h>

// WaveletTransform (à trous, db4, 4 levels) for MI455X / gfx1250.
//
// Each dense filter row has 8 non-zeros at dilated positions (m - 2^j*i) mod
// 2048, so the reference's 8 dense 512x2048x2048 GEMMs are really chained
// dilated circular convolutions. Per 16x16 output tile only 72 virtual K
// columns matter, and the 4x16 WMMA B-tiles depend only on (kk, lane) — not
// the tile position — so B is synthesized from the 8 taps (zero-padded LDS
// table, taps read from row 0 of the dense filter inputs) while A streams
// from the activations. Matrix math = V_WMMA_F32_16X16X4_F32, full fp32.
//
// Level index J is a template parameter: band constants fold, the K loop
// fully unrolls (6/8/11/18 steps x 2 WMMAs), and the last-level / level-0
// specializations remove all epilogue branching.
//
// Traffic ~40MB (x 4MB + out 20MB + 2x4MB ws ping-pong, L2-resident)
// => ~2us at 23.3 TB/s, vs ~134MB filter reads + 34 GFLOP dense.

typedef __attribute__((ext_vector_type(2))) float v2f;
typedef __attribute__((ext_vector_type(8))) float v8f;

#define WT_L 2048
#define WT_D 32
#define WT_BD 512          // B*D = 16*32
#define WT_NBLK 32         // BD / 16
#define WT_MTILES 128      // L / 16

template <int J, bool IS_X, bool LAST>
__global__ __launch_bounds__(256) void wavelet_level_kernel(
    const float* __restrict__ in,     // activations: x (level 0) or ws buffer
    const float* __restrict__ lo_f,   // v_dec_filter[J] plane (row 0 used)
    const float* __restrict__ hi_f,   // w_dec_filter[J] plane (row 0 used)
    float* __restrict__ ws_out,       // v_J destination (unused when LAST)
    float* __restrict__ out)          // (B, D, L, 5) fp32
{
    // Zero-padded tap tables in LDS (slots 8..15 = 0 -> invalid lanes read 0).
    __shared__ float s_lo[16];
    __shared__ float s_hi[16];
    const int tid = threadIdx.x;
    if (tid < 16) {
        float lo = 0.0f, hi = 0.0f;
        if (tid < 8) {
            // filter[J][t=0][(0 - 2^J * i) mod L] == tap[i]
            const int col = (WT_L - (tid << J)) & (WT_L - 1);
            lo = lo_f[col];
            hi = hi_f[col];
        }
        s_lo[tid] = lo;
        s_hi[tid] = hi;
    }
    __syncthreads();

    // One wave per (bd-block, m-tile): 32 * 128 = 4096 waves.
    const int wave  = (blockIdx.x << 3) + (tid >> 5);
    const int lane  = tid & 31;
    const int blk   = wave >> 7;             // 0..31: which 16 bd-rows
    const int m0    = (wave & 127) << 4;     // 0..2032: output column tile
    const int n     = lane & 15;             // N (and A's M) index
    const int khalf = lane >> 4;             // 0/1
    const int k0    = khalf << 1;            // K base for this half-wave

    // Level 0 reads x (B,L,D) transpose-on-load (lanes walk contiguous d =>
    // coalesced); later levels read ws laid out [blk][l][16].
    constexpr int LSTRIDE = IS_X ? WT_D : 16;
    const float* inp = IS_X
        ? in + (unsigned)(blk >> 1) * (WT_L * WT_D) + ((blk & 1) << 4) + n
        : in + (unsigned)blk * (WT_L * 16) + n;

    v8f acc_lo = {};  // -> v_J
    v8f acc_hi = {};  // -> w_J
    // First K-step whose B tile can be nonzero: kk >= 53 - 7*2^J.
    constexpr int KK0raw = 53 - (7 << J);
    constexpr int KK0    = (KK0raw < 0) ? 0 : ((KK0raw + 3) & ~3);
    constexpr int JMASK  = (1 << J) - 1;

#pragma unroll
    for (int kk = KK0; kk <= 68; kk += 4) {
        v2f a, blo, bhi;
#pragma unroll
        for (int e = 0; e < 2; ++e) {
            const int kc = k0 + e;                 // this lane's K index
            // A[M=n][K=kc] = in[(m0 - 56 + kk + kc) mod L][n]
            const int lv = m0 - 56 + kk + kc;      // virtual column, >= -56
            const int l  = (lv + WT_L) & (WT_L - 1);
            a[e] = inp[l * LSTRIDE];
            // B[K=kc][N=n] = tap[delta>>J] iff delta = n+56-kk-kc is a valid
            // dilated offset; else reads zero slot.
            const int delta = n + 56 - kk - kc;
            const int i = delta >> J;
            const bool valid = (delta >= 0) && ((delta & JMASK) == 0) && (i < 8);
            const int idx = valid ? i : 15;
            blo[e] = s_lo[idx];
            bhi[e] = s_hi[idx];
        }
        acc_lo = __builtin_amdgcn_wmma_f32_16x16x4_f32(
            false, a, false, blo, (short)0, acc_lo, false, false);
        acc_hi = __builtin_amdgcn_wmma_f32_16x16x4_f32(
            false, a, false, bhi, (short)0, acc_hi, false, false);
    }

    // D layout: VGPR r holds D[M = r + 8*khalf][N = n].
    const int Mbase = khalf << 3;
    const int m = m0 + n;
    // w_J -> output slot J; one base address + immediate offsets (r*L*5).
    const unsigned obase = ((unsigned)((blk << 4) + Mbase) * WT_L + m) * 5u + J;
#pragma unroll
    for (int r = 0; r < 8; ++r)
        out[obase + (unsigned)r * (WT_L * 5)] = acc_hi[r];

    if constexpr (!LAST) {
        // v_J -> ping-pong buffer [blk][m][16]; 8 contiguous floats per lane
        // => two b128 stores (32B-aligned).
        float4* wp = (float4*)(ws_out +
            (unsigned)blk * (WT_L * 16) + (unsigned)m * 16 + Mbase);
        wp[0] = make_float4(acc_lo[0], acc_lo[1], acc_lo[2], acc_lo[3]);
        wp[1] = make_float4(acc_lo[4], acc_lo[5], acc_lo[6], acc_lo[7]);
    } else {
        // last level: v_3 -> output slot 4 (obase points at slot J==3 => +1).
        const unsigned obase2 = obase + (4 - J);
#pragma unroll
        for (int r = 0; r < 8; ++r)
            out[obase2 + (unsigned)r * (WT_L * 5)] = acc_lo[r];
    }
}

extern "C" void kernel_launch(void* const* d_in, const int* in_sizes, int n_in,
                              void* d_out, int out_size, void* d_ws, size_t ws_size,
                              hipStream_t stream) {
    (void)in_sizes; (void)n_in; (void)out_size; (void)ws_size;
    const float* x  = (const float*)d_in[0];   // (16, 2048, 32) fp32
    const float* wf = (const float*)d_in[1];   // w_dec_filter (hi), (4, L, L)
    const float* vf = (const float*)d_in[2];   // v_dec_filter (lo), (4, L, L)
    float* out  = (float*)d_out;               // (16, 32, 2048, 5) fp32
    float* buf0 = (float*)d_ws;                // 512*2048 fp32 = 4 MB
    float* buf1 = buf0 + (size_t)WT_BD * WT_L; // second 4 MB (needs ws >= 8 MB)

    const dim3 grid(WT_NBLK * WT_MTILES / 8);  // 512 blocks, 8 waves each
    const dim3 block(256);
    const size_t fs = (size_t)WT_L * WT_L;     // per-level filter plane

    wavelet_level_kernel<0, true,  false><<<grid, block, 0, stream>>>(
        x,    vf,          wf,          buf0,    out);
    wavelet_level_kernel<1, false, false><<<grid, block, 0, stream>>>(
        buf0, vf + fs,     wf + fs,     buf1,    out);
    wavelet_level_kernel<2, false, false><<<grid, block, 0, stream>>>(
        buf1, vf + 2 * fs, wf + 2 * fs, buf0,    out);
    wavelet_level_kernel<3, false, true ><<<grid, block, 0, stream>>>(
        buf0, vf + 3 * fs, wf + 3 * fs, nullptr, out);
}